// SAGEConv_1898375544833
// MI455X (gfx1250) — compile-verified
//
#include <hip/hip_runtime.h>
#include <hip/hip_bf16.h>

typedef __attribute__((ext_vector_type(2))) float v2f;
typedef __attribute__((ext_vector_type(8))) float v8f;

#define D_FEAT 64

// ---------------------------------------------------------------------------
// Kernel 1: zero the workspace accumulators (agg[N*64] ++ deg[N])
// ---------------------------------------------------------------------------
__global__ void sage_zero_kernel(float* __restrict__ ws, long n) {
    long i = (long)blockIdx.x * blockDim.x + threadIdx.x;
    long stride = (long)gridDim.x * blockDim.x;
    for (; i < n; i += stride) ws[i] = 0.0f;
}

// ---------------------------------------------------------------------------
// Kernel 2: edge scatter. 16 lanes per edge, float4 per lane (64 floats/row).
// agg[dst] += feat[src] via f32 atomics (L2-resident, 25.6MB target);
// deg[dst] += 1 from lane 0 of each edge group.
// ---------------------------------------------------------------------------
__global__ __launch_bounds__(256) void sage_scatter_kernel(
        const float4* __restrict__ feat4,
        const int* __restrict__ src,
        const int* __restrict__ dst,
        float* __restrict__ agg,
        float* __restrict__ deg,
        int E) {
    int t = blockIdx.x * blockDim.x + threadIdx.x;
    int e = t >> 4;
    int lane16 = t & 15;
    if (e >= E) return;
    int s = src[e];
    int d = dst[e];
    float4 v = feat4[(long)s * (D_FEAT / 4) + lane16];
    float* o = agg + (long)d * D_FEAT + lane16 * 4;
    atomicAdd(o + 0, v.x);
    atomicAdd(o + 1, v.y);
    atomicAdd(o + 2, v.z);
    atomicAdd(o + 3, v.w);
    if (lane16 == 0) atomicAdd(deg + d, 1.0f);
}

// ---------------------------------------------------------------------------
// Kernel 3: out = feat @ W_self^T + (agg/max(deg,1)) @ W_neigh^T
// fp32 WMMA (V_WMMA_F32_16X16X4_F32). One wave per 16-row tile, 4 column
// tiles (D_OUT=64) accumulated simultaneously; both matmuls fused into the
// same accumulators. N % 16 == 0, so EXEC is always all-1s inside WMMA path.
//
// A (16x4 f32) layout: lane L -> row M=L%16, holds K = k0 + 2*(L>>4) + {0,1}
// B (4x16 f32) layout: lane L -> col N=L%16, holds K = k0 + 2*(L>>4) + {0,1}
//   with B[k][j] = W[j][k]  (W is [64,64] row-major, out = X @ W^T)
// C/D (16x16 f32): VGPR v, lane L -> (M = v + 8*(L>>4), N = L%16)
// ---------------------------------------------------------------------------
__global__ __launch_bounds__(256) void sage_wmma_kernel(
        const float* __restrict__ feat,
        const float* __restrict__ agg,
        const float* __restrict__ deg,
        const float* __restrict__ Wself,
        const float* __restrict__ Wneigh,
        float* __restrict__ out,
        int N) {
    const int lane = threadIdx.x & 31;   // wave32
    const int wave = threadIdx.x >> 5;   // 0..7
    const int m    = lane & 15;          // row/col within 16-tile
    const int hh   = lane >> 4;          // half-wave select (K pairing)

    const int row_base = (blockIdx.x * 8 + wave) * 16;
    if (row_base >= N) return;           // wave-uniform: EXEC stays all-1s

    const int row = row_base + m;
    const float dg  = deg[row];
    const float inv = 1.0f / fmaxf(dg, 1.0f);

    const float* frow = feat + (long)row * D_FEAT + 2 * hh;
    const float* arow = agg  + (long)row * D_FEAT + 2 * hh;

    v8f acc[4];
#pragma unroll
    for (int ct = 0; ct < 4; ++ct) acc[ct] = (v8f){};

#pragma unroll
    for (int s = 0; s < 16; ++s) {
        const int k0 = 4 * s;
        v2f a_s = *(const v2f*)(frow + k0);
        v2f a_n = *(const v2f*)(arow + k0);
        a_n = a_n * inv;   // mean aggregation applied on the fly
#pragma unroll
        for (int ct = 0; ct < 4; ++ct) {
            const int j = ct * 16 + m;   // output column held by this lane
            v2f b_s = *(const v2f*)(Wself  + (long)j * D_FEAT + 2 * hh + k0);
            v2f b_n = *(const v2f*)(Wneigh + (long)j * D_FEAT + 2 * hh + k0);
            acc[ct] = __builtin_amdgcn_wmma_f32_16x16x4_f32(
                false, a_s, false, b_s, (short)0, acc[ct], false, false);
            acc[ct] = __builtin_amdgcn_wmma_f32_16x16x4_f32(
                false, a_n, false, b_n, (short)0, acc[ct], false, false);
        }
    }

    // Store D tiles (coalesced across lanes along the column dimension).
#pragma unroll
    for (int ct = 0; ct < 4; ++ct) {
#pragma unroll
        for (int v = 0; v < 8; ++v) {
            const int r = row_base + v + 8 * hh;
            out[(long)r * D_FEAT + ct * 16 + m] = acc[ct][v];
        }
    }
}

// ---------------------------------------------------------------------------
extern "C" void kernel_launch(void* const* d_in, const int* in_sizes, int n_in,
                              void* d_out, int out_size, void* d_ws, size_t ws_size,
                              hipStream_t stream) {
    const float* feat   = (const float*)d_in[0];
    const int*   src    = (const int*)d_in[1];
    const int*   dst    = (const int*)d_in[2];
    const float* Wself  = (const float*)d_in[3];
    const float* Wneigh = (const float*)d_in[4];
    float* out = (float*)d_out;

    const int N = in_sizes[0] / D_FEAT;   // 100000
    const int E = in_sizes[1];            // 1600000

    // Workspace layout: agg[N*64] floats, then deg[N] floats (~26 MB).
    float* agg = (float*)d_ws;
    float* deg = agg + (size_t)N * D_FEAT;

    // 1) zero accumulators
    const long zn = (long)N * (D_FEAT + 1);
    sage_zero_kernel<<<2048, 256, 0, stream>>>((float*)d_ws, zn);

    // 2) edge scatter: 16 threads per edge
    const long sth = (long)E * 16;
    const int sblocks = (int)((sth + 255) / 256);
    sage_scatter_kernel<<<sblocks, 256, 0, stream>>>(
        (const float4*)feat, src, dst, agg, deg, E);

    // 3) fused dual-GEMM with fp32 WMMA: 128 rows per block (8 waves x 16)
    const int gblocks = (N + 127) / 128;
    sage_wmma_kernel<<<gblocks, 256, 0, stream>>>(
        feat, agg, deg, Wself, Wneigh, out, N);
}